// RobertaBasedDecoder_64561948393798
// MI455X (gfx1250) — compile-verified
//
#include <hip/hip_runtime.h>
#include <hip/hip_bf16.h>

#define EMB 768
#define Hh  384
#define Bb  16
#define Tt  128
#define Vv  50265
#define APITCH 770        // LDS row pitch (even -> b64-aligned, kills bank conflicts)

typedef __attribute__((ext_vector_type(2))) float v2f;
typedef __attribute__((ext_vector_type(4))) float v4f;
typedef __attribute__((ext_vector_type(8))) float v8f;
typedef int v4i __attribute__((vector_size(4 * sizeof(int))));

#if defined(__HIP_DEVICE_COMPILE__) && __has_builtin(__builtin_amdgcn_global_load_async_to_lds_b128)
#define HAVE_ASYNC_LDS 1
typedef __attribute__((address_space(1))) v4i* g4p;   // global int4*
typedef __attribute__((address_space(3))) v4i* l4p;   // LDS int4*
#else
#define HAVE_ASYNC_LDS 0
#endif

__device__ __forceinline__ float sigmoidf_(float x) {
    return 1.0f / (1.0f + __expf(-x));
}

__device__ __forceinline__ void wait_async_then_barrier() {
#if HAVE_ASYNC_LDS
#if __has_builtin(__builtin_amdgcn_s_wait_asynccnt)
    __builtin_amdgcn_s_wait_asynccnt(0);
#else
    asm volatile("s_wait_asynccnt 0" ::: "memory");
#endif
#endif
    __syncthreads();
}

// ---------------------------------------------------------------------------
// Init: build xs[t][b][e] = (t==0 ? GO : target[b][t-1][e]); copy encoder
// states into ping-pong buffer 0.
// ---------------------------------------------------------------------------
__global__ __launch_bounds__(256) void init_kernel(
    const float* __restrict__ target, const float* __restrict__ eh,
    const float* __restrict__ ec, float* __restrict__ xs,
    float* __restrict__ hb, float* __restrict__ cb)
{
    const int i = blockIdx.x * 256 + threadIdx.x;
    const int N1 = Tt * Bb * EMB;
    if (i < N1) {
        int t = i / (Bb * EMB);
        int r = i % (Bb * EMB);
        int b = r / EMB;
        int e = r % EMB;
        xs[i] = (t == 0) ? -1.0f : target[((size_t)b * Tt + (t - 1)) * EMB + e];
    } else {
        int j = i - N1;
        if (j < 4 * Bb * Hh)            hb[j] = eh[j];
        else if (j < 8 * Bb * Hh)       cb[j - 4 * Bb * Hh] = ec[j - 4 * Bb * Hh];
    }
}

// ---------------------------------------------------------------------------
// One LSTM step for one layer (both directional cells).
// 48 waves: wave w -> dir d = w/24, unit tile u0 = (w%24)*16.
// Each wave accumulates 4 gate tiles (i,f,g,o) for its 16-unit column slice
// with V_WMMA_F32_16X16X4_F32, then does the fused gate math in-register.
// ---------------------------------------------------------------------------
__global__ __launch_bounds__(64) void lstm_step_kernel(
    const float* __restrict__ x,      // 16 x 768 layer input (row stride EMB)
    const float* __restrict__ hr,     // read  h states (4,B,H)
    const float* __restrict__ cr,     // read  c states (4,B,H)
    float* __restrict__ hw,           // write h states (4,B,H)
    float* __restrict__ cw,           // write c states (4,B,H)
    const float* __restrict__ Wih,    // (2,2,4H,EMB)
    const float* __restrict__ Whh,    // (2,2,4H,H)
    const float* __restrict__ bih,    // (2,2,4H)
    const float* __restrict__ bhh,    // (2,2,4H)
    float* __restrict__ dst,          // 16 x 768 concat output for this t
    int l)
{
    const int lane = threadIdx.x & 31;
    const int w    = blockIdx.x * 2 + (threadIdx.x >> 5);
    const int d    = w / 24;
    const int u0   = (w % 24) * 16;
    const int idx  = 2 * l + d;
    const int n    = lane & 15;      // column within tile / A row (dual role)
    const int g    = lane >> 4;
    const int k2   = 2 * g;

    const float* Wx = Wih + (size_t)idx * (4 * Hh) * EMB;
    const float* Wh = Whh + (size_t)idx * (4 * Hh) * Hh;
    const float* Ah = hr + idx * (Bb * Hh);

    v8f a0 = {}, a1 = {}, a2 = {}, a3 = {};

    // x @ Wih.T  (K = 768)
    {
        const float* ab = x + n * EMB + k2;
        const float* b0 = Wx + (size_t)(0 * Hh + u0 + n) * EMB + k2;
        const float* b1 = Wx + (size_t)(1 * Hh + u0 + n) * EMB + k2;
        const float* b2 = Wx + (size_t)(2 * Hh + u0 + n) * EMB + k2;
        const float* b3 = Wx + (size_t)(3 * Hh + u0 + n) * EMB + k2;
        #pragma unroll 4
        for (int k = 0; k < EMB; k += 4) {
            v2f a = *(const v2f*)(ab + k);
            a0 = __builtin_amdgcn_wmma_f32_16x16x4_f32(false, a, false, *(const v2f*)(b0 + k), (short)0, a0, false, false);
            a1 = __builtin_amdgcn_wmma_f32_16x16x4_f32(false, a, false, *(const v2f*)(b1 + k), (short)0, a1, false, false);
            a2 = __builtin_amdgcn_wmma_f32_16x16x4_f32(false, a, false, *(const v2f*)(b2 + k), (short)0, a2, false, false);
            a3 = __builtin_amdgcn_wmma_f32_16x16x4_f32(false, a, false, *(const v2f*)(b3 + k), (short)0, a3, false, false);
        }
    }
    // h @ Whh.T  (K = 384)
    {
        const float* ab = Ah + n * Hh + k2;
        const float* b0 = Wh + (size_t)(0 * Hh + u0 + n) * Hh + k2;
        const float* b1 = Wh + (size_t)(1 * Hh + u0 + n) * Hh + k2;
        const float* b2 = Wh + (size_t)(2 * Hh + u0 + n) * Hh + k2;
        const float* b3 = Wh + (size_t)(3 * Hh + u0 + n) * Hh + k2;
        #pragma unroll 4
        for (int k = 0; k < Hh; k += 4) {
            v2f a = *(const v2f*)(ab + k);
            a0 = __builtin_amdgcn_wmma_f32_16x16x4_f32(false, a, false, *(const v2f*)(b0 + k), (short)0, a0, false, false);
            a1 = __builtin_amdgcn_wmma_f32_16x16x4_f32(false, a, false, *(const v2f*)(b1 + k), (short)0, a1, false, false);
            a2 = __builtin_amdgcn_wmma_f32_16x16x4_f32(false, a, false, *(const v2f*)(b2 + k), (short)0, a2, false, false);
            a3 = __builtin_amdgcn_wmma_f32_16x16x4_f32(false, a, false, *(const v2f*)(b3 + k), (short)0, a3, false, false);
        }
    }

    const int col = u0 + n;                        // unit index in [0, 384)
    const int bo  = idx * 4 * Hh;
    const float bi_ = bih[bo + 0 * Hh + col] + bhh[bo + 0 * Hh + col];
    const float bf_ = bih[bo + 1 * Hh + col] + bhh[bo + 1 * Hh + col];
    const float bg_ = bih[bo + 2 * Hh + col] + bhh[bo + 2 * Hh + col];
    const float bo_ = bih[bo + 3 * Hh + col] + bhh[bo + 3 * Hh + col];

    #pragma unroll
    for (int r = 0; r < 8; ++r) {
        const int m = r + 8 * g;                   // batch row
        float gi = sigmoidf_(a0[r] + bi_);
        float gf = sigmoidf_(a1[r] + bf_);
        float gc = tanhf(a2[r] + bg_);
        float go = sigmoidf_(a3[r] + bo_);
        float cold = cr[idx * Bb * Hh + m * Hh + col];
        float cn = gf * cold + gi * gc;
        float hn = go * tanhf(cn);
        cw[idx * Bb * Hh + m * Hh + col] = cn;
        hw[idx * Bb * Hh + m * Hh + col] = hn;
        dst[m * EMB + d * Hh + col] = hn;          // concat into layer output
    }
}

// ---------------------------------------------------------------------------
// Batched vocabulary projection: out[b][t][v] = y[t][b] . Wc[v] + bc[v].
// TWO timesteps per block (M-reuse): both 16x768 A-panels staged in LDS
// (98.5 KB -- legal on CDNA5's 320KB/WGP), preferentially via
// GLOBAL_LOAD_ASYNC_TO_LDS_B128 (ASYNCcnt). Each wave owns one 16-col Wc
// tile and accumulates two C tiles against a single B load stream, halving
// Wc/L2 traffic versus one-timestep tiles.
// ---------------------------------------------------------------------------
__global__ __launch_bounds__(256) void logits_kernel(
    const float* __restrict__ y, const float* __restrict__ Wc,
    const float* __restrict__ bc, float* __restrict__ out)
{
    __shared__ __align__(16) float As[2][16 * APITCH];
    const int t0  = blockIdx.y * 2;
    const int tid = threadIdx.x;

    #pragma unroll
    for (int p = 0; p < 2; ++p) {
        const float* src = y + (size_t)(t0 + p) * Bb * EMB;
        for (int i = tid * 4; i < Bb * EMB; i += 256 * 4) {
            int rrow = i / EMB, cc = i % EMB;     // 16B chunks never straddle rows
#if HAVE_ASYNC_LDS
            __builtin_amdgcn_global_load_async_to_lds_b128(
                (g4p)(src + i), (l4p)&As[p][rrow * APITCH + cc], 0, 0);
#else
            *(v4f*)&As[p][rrow * APITCH + cc] = *(const v4f*)(src + i);
#endif
        }
    }
    wait_async_then_barrier();

    const int lane  = tid & 31;
    const int ntile = blockIdx.x * 8 + (tid >> 5);
    if (ntile * 16 >= Vv) return;

    const int n = lane & 15, g = lane >> 4;
    const int col  = ntile * 16 + n;
    const int colc = (col < Vv) ? col : (Vv - 1);

    const float* wrow = Wc + (size_t)colc * EMB + 2 * g;
    const float* ar0  = &As[0][n * APITCH + 2 * g];
    const float* ar1  = &As[1][n * APITCH + 2 * g];

    v8f acc0 = {}, acc1 = {};
    #pragma unroll 4
    for (int k = 0; k < EMB; k += 4) {
        v2f b  = *(const v2f*)(wrow + k);
        v2f x0 = *(const v2f*)(ar0 + k);
        v2f x1 = *(const v2f*)(ar1 + k);
        acc0 = __builtin_amdgcn_wmma_f32_16x16x4_f32(false, x0, false, b, (short)0, acc0, false, false);
        acc1 = __builtin_amdgcn_wmma_f32_16x16x4_f32(false, x1, false, b, (short)0, acc1, false, false);
    }

    const float bias = bc[colc];
    if (col < Vv) {
        #pragma unroll
        for (int r = 0; r < 8; ++r) {
            const int m = r + 8 * g;               // batch row
            out[(size_t)m * ((size_t)Tt * Vv) + (size_t)(t0 + 0) * Vv + col] = acc0[r] + bias;
            out[(size_t)m * ((size_t)Tt * Vv) + (size_t)(t0 + 1) * Vv + col] = acc1[r] + bias;
        }
    }
}

// ---------------------------------------------------------------------------
// Copy final states (ping-pong buffer 0 after 128 steps) to output tail.
// ---------------------------------------------------------------------------
__global__ __launch_bounds__(256) void finalize_kernel(
    const float* __restrict__ hb, const float* __restrict__ cb, float* __restrict__ o)
{
    const int i = blockIdx.x * 256 + threadIdx.x;   // grid covers 2*4*B*H exactly
    if (i < 4 * Bb * Hh) o[i] = hb[i];
    else                 o[i] = cb[i - 4 * Bb * Hh];
}

extern "C" void kernel_launch(void* const* d_in, const int* in_sizes, int n_in,
                              void* d_out, int out_size, void* d_ws, size_t ws_size,
                              hipStream_t stream)
{
    const float* enc_h  = (const float*)d_in[0];
    const float* enc_c  = (const float*)d_in[1];
    const float* target = (const float*)d_in[2];
    const float* W_ih   = (const float*)d_in[3];
    const float* W_hh   = (const float*)d_in[4];
    const float* b_ih   = (const float*)d_in[5];
    const float* b_hh   = (const float*)d_in[6];
    const float* Wc     = (const float*)d_in[7];
    const float* bc     = (const float*)d_in[8];
    float* out = (float*)d_out;

    float* ws   = (float*)d_ws;
    float* xs   = ws;                           // T*B*EMB
    float* hcat = xs   + (size_t)Tt * Bb * EMB; // layer-0 concat per t
    float* yb   = hcat + (size_t)Tt * Bb * EMB; // layer-1 concat per t
    float* hbuf = yb   + (size_t)Tt * Bb * EMB; // 2 * (4*B*H) ping-pong
    float* cbuf = hbuf + 2 * 4 * Bb * Hh;

    const int SB = 4 * Bb * Hh;

    init_kernel<<<6336, 256, 0, stream>>>(target, enc_h, enc_c, xs, hbuf, cbuf);

    for (int t = 0; t < Tt; ++t) {
        float* hr = hbuf + (t & 1) * SB;
        float* hw = hbuf + ((t + 1) & 1) * SB;
        float* cr = cbuf + (t & 1) * SB;
        float* cw = cbuf + ((t + 1) & 1) * SB;
        lstm_step_kernel<<<24, 64, 0, stream>>>(xs + (size_t)t * Bb * EMB, hr, cr, hw, cw,
                                                W_ih, W_hh, b_ih, b_hh,
                                                hcat + (size_t)t * Bb * EMB, 0);
        lstm_step_kernel<<<24, 64, 0, stream>>>(hcat + (size_t)t * Bb * EMB, hr, cr, hw, cw,
                                                W_ih, W_hh, b_ih, b_hh,
                                                yb + (size_t)t * Bb * EMB, 1);
    }

    dim3 lg((((Vv + 15) / 16) + 7) / 8, Tt / 2);   // (393, 64)
    logits_kernel<<<lg, 256, 0, stream>>>(yb, Wc, bc, out);

    finalize_kernel<<<192, 256, 0, stream>>>(hbuf, cbuf, out + (size_t)Bb * Tt * Vv);
}